// CausalSelfAttention_37503654428993
// MI455X (gfx1250) — compile-verified
//
#include <hip/hip_runtime.h>
#include <hip/hip_bf16.h>
#include <math.h>

// Causal self-attention, fp32, using V_WMMA_F32_16X16X4_F32 (CDNA5 / gfx1250).
// Bandwidth-bound (~70 GFLOP vs ~350MB of HBM traffic) -> keep fp32 precision;
// exploit causal zeros to halve the dominant weights@V GEMM; share the A
// operand across two N-tiles per wave. Wave-uniform control values are forced
// into SGPRs via readfirstlane so loops/branches are scalar (WMMA needs
// EXEC == all ones; scalar control flow guarantees it structurally).

typedef __attribute__((ext_vector_type(2))) float v2f;
typedef __attribute__((ext_vector_type(8))) float v8f;

// ---------------------------------------------------------------------------
// C[M,N] = A[M,K] * B[K,N], all row-major fp32.
// Block = 128 threads = 4 waves (2x2); block tile 32x64, wave tile 16x32
// (two 16x16 accumulators sharing one A operand).
// If `causal` != 0, A's columns k > row are known-zero, so the K loop for a
// wave covering rows [row0, row0+15] stops at row0+16 (scalarized bound).
// Batch strides applied via blockIdx.z.
// ---------------------------------------------------------------------------
__global__ __launch_bounds__(128)
void gemm_nn_f32(const float* __restrict__ A, const float* __restrict__ B,
                 float* __restrict__ C, int N, int K,
                 long sA, long sB, long sC, int causal) {
  A += (long)blockIdx.z * sA;
  B += (long)blockIdx.z * sB;
  C += (long)blockIdx.z * sC;

  const int wave = threadIdx.x >> 5;
  const int lane = threadIdx.x & 31;
  const int lh   = lane & 15;      // lane within half-wave
  const int hi   = lane >> 4;      // half select (+2 on K, +8 on M-of-C)

  const int row0 = blockIdx.y * 32 + (wave >> 1) * 16;
  const int col0 = blockIdx.x * 64 + (wave & 1) * 32;

  const float* Ap  = A + (size_t)(row0 + lh) * K + hi * 2;     // A[row][kk+hi*2 ...]
  const float* Bp0 = B + (size_t)(hi * 2) * N + col0 + lh;     // B[kk+hi*2][col]
  const float* Bp1 = Bp0 + 16;

  // Wave-uniform loop bound, forced scalar (SGPR) -> scalar loop control.
  const int Kend =
      __builtin_amdgcn_readfirstlane(causal ? min(K, row0 + 16) : K);

  v8f acc0 = {};
  v8f acc1 = {};
#pragma unroll 4
  for (int kk = 0; kk < Kend; kk += 4) {
    v2f a = *(const v2f*)(Ap + kk);            // A[row][kk+hi*2+0], +1
    v2f b0, b1;
    b0.x = Bp0[(size_t)kk * N];
    b0.y = Bp0[(size_t)kk * N + N];
    b1.x = Bp1[(size_t)kk * N];
    b1.y = Bp1[(size_t)kk * N + N];
    acc0 = __builtin_amdgcn_wmma_f32_16x16x4_f32(
        false, a, false, b0, (short)0, acc0, false, false);
    acc1 = __builtin_amdgcn_wmma_f32_16x16x4_f32(
        false, a, false, b1, (short)0, acc1, false, false);
  }

  float* Cp = C + (size_t)(row0 + hi * 8) * N + col0 + lh;
#pragma unroll
  for (int r = 0; r < 8; ++r) {
    Cp[(size_t)r * N]      = acc0[r];
    Cp[(size_t)r * N + 16] = acc1[r];
  }
}

// ---------------------------------------------------------------------------
// S[q,j] = (Q[q,:] dot Km[j,:]) * scale for j <= q, else -1e30 (causal).
// NT GEMM: A = Q [T,DA], B^T = Km [T,DA]; per-batch via blockIdx.z.
// Wave tile 16x32 (two accumulators sharing the Q operand).
// ---------------------------------------------------------------------------
__global__ __launch_bounds__(128)
void scores_nt_causal(const float* __restrict__ Q, const float* __restrict__ Km,
                      float* __restrict__ S, int T, int DA, float scale) {
  Q  += (size_t)blockIdx.z * T * DA;
  Km += (size_t)blockIdx.z * T * DA;
  S  += (size_t)blockIdx.z * T * T;

  const int wave = threadIdx.x >> 5;
  const int lane = threadIdx.x & 31;
  const int lh   = lane & 15;
  const int hi   = lane >> 4;

  const int row0 = blockIdx.y * 32 + (wave >> 1) * 16;
  const int col0 = blockIdx.x * 64 + (wave & 1) * 32;

  const int col = col0 + lh;
  float* Sp = S + (size_t)(row0 + hi * 8) * T + col;

  // Wave tile entirely above the diagonal: scalar branch (readfirstlane),
  // no WMMA, just write the mask value.
  if (__builtin_amdgcn_readfirstlane(col0 > row0 + 15 ? 1 : 0)) {
#pragma unroll
    for (int r = 0; r < 8; ++r) {
      Sp[(size_t)r * T]      = -1.0e30f;
      Sp[(size_t)r * T + 16] = -1.0e30f;
    }
    return;
  }

  const float* Qp  = Q + (size_t)(row0 + lh) * DA + hi * 2;        // A operand
  const float* Kp0 = Km + (size_t)(col0 + lh) * DA + hi * 2;       // B^T (contiguous)
  const float* Kp1 = Kp0 + (size_t)16 * DA;

  v8f acc0 = {};
  v8f acc1 = {};
#pragma unroll 4
  for (int kk = 0; kk < DA; kk += 4) {
    v2f a  = *(const v2f*)(Qp + kk);
    v2f b0 = *(const v2f*)(Kp0 + kk);
    v2f b1 = *(const v2f*)(Kp1 + kk);
    acc0 = __builtin_amdgcn_wmma_f32_16x16x4_f32(
        false, a, false, b0, (short)0, acc0, false, false);
    acc1 = __builtin_amdgcn_wmma_f32_16x16x4_f32(
        false, a, false, b1, (short)0, acc1, false, false);
  }

#pragma unroll
  for (int r = 0; r < 8; ++r) {
    const int row = row0 + hi * 8 + r;
    const float v0 = acc0[r] * scale;
    const float v1 = acc1[r] * scale;
    Sp[(size_t)r * T]      = (col <= row)      ? v0 : -1.0e30f;
    Sp[(size_t)r * T + 16] = (col + 16 <= row) ? v1 : -1.0e30f;
  }
}

// ---------------------------------------------------------------------------
// In-place row softmax over rows of length T. One 256-thread block per row.
// Masked entries are -1e30 -> exp() underflows to exactly 0.
// ---------------------------------------------------------------------------
__global__ __launch_bounds__(256)
void softmax_rows(float* __restrict__ W, int T) {
  float* p = W + (size_t)blockIdx.x * T;
  __shared__ float red[256];
  const int t = threadIdx.x;

  float m = -3.0e38f;
  for (int i = t; i < T; i += 256) m = fmaxf(m, p[i]);
  red[t] = m;
  __syncthreads();
  for (int s = 128; s > 0; s >>= 1) {
    if (t < s) red[t] = fmaxf(red[t], red[t + s]);
    __syncthreads();
  }
  m = red[0];
  __syncthreads();

  float sum = 0.0f;
  for (int i = t; i < T; i += 256) sum += __expf(p[i] - m);
  red[t] = sum;
  __syncthreads();
  for (int s = 128; s > 0; s >>= 1) {
    if (t < s) red[t] += red[t + s];
    __syncthreads();
  }
  const float inv = 1.0f / red[0];

  for (int i = t; i < T; i += 256) p[i] = __expf(p[i] - m) * inv;
}

// ---------------------------------------------------------------------------
// Launch
// ---------------------------------------------------------------------------
extern "C" void kernel_launch(void* const* d_in, const int* in_sizes, int n_in,
                              void* d_out, int out_size, void* d_ws, size_t ws_size,
                              hipStream_t stream) {
  (void)in_sizes; (void)n_in; (void)out_size; (void)ws_size;

  const int B = 4, T = 2048, D = 1024, DA = 256;

  const float* x  = (const float*)d_in[0];   // [B,T,D]
  const float* Wq = (const float*)d_in[1];   // [D,DA]
  const float* Wk = (const float*)d_in[2];   // [D,DA]
  const float* Wv = (const float*)d_in[3];   // [D,D]

  float* out_attn = (float*)d_out;                            // [B,T,D]
  float* out_w    = (float*)d_out + (size_t)B * T * D;        // [B,T,T]

  // Workspace: Q (8MB) | K (8MB) | V (32MB) = 48MB total.
  float* Qws = (float*)d_ws;
  float* Kws = Qws + (size_t)B * T * DA;
  float* Vws = Kws + (size_t)B * T * DA;

  const dim3 blk(128);

  // 1) Projections: [B*T, D] x [D, {DA,DA,D}]   (block tile 32x64)
  gemm_nn_f32<<<dim3(DA / 64, (B * T) / 32, 1), blk, 0, stream>>>(
      x, Wq, Qws, DA, D, 0, 0, 0, 0);
  gemm_nn_f32<<<dim3(DA / 64, (B * T) / 32, 1), blk, 0, stream>>>(
      x, Wk, Kws, DA, D, 0, 0, 0, 0);
  gemm_nn_f32<<<dim3(D / 64, (B * T) / 32, 1), blk, 0, stream>>>(
      x, Wv, Vws, D, D, 0, 0, 0, 0);

  // 2) Causal scores: per-batch [T,T] = Q K^T / sqrt(DA)
  scores_nt_causal<<<dim3(T / 64, T / 32, B), blk, 0, stream>>>(
      Qws, Kws, out_w, T, DA, 1.0f / 16.0f);

  // 3) Row softmax over all B*T rows
  softmax_rows<<<dim3(B * T), dim3(256), 0, stream>>>(out_w, T);

  // 4) attn_output = weights @ V (batched over B; causal -> K loop clipped
  //    to row0+16, halving FLOPs and weight reads of the dominant GEMM)
  gemm_nn_f32<<<dim3(D / 64, T / 32, B), blk, 0, stream>>>(
      out_w, Vws, out_attn, D, T,
      (long)T * T, (long)T * D, (long)T * D, 1);
}